// GreedyFeatureInit_24446953848979
// MI455X (gfx1250) — compile-verified
//
#include <hip/hip_runtime.h>
#include <math.h>

// GreedyFeatureInit on MI455X (gfx1250, wave32).
// One workgroup (256 threads = 8 waves) per batch element; all 16 greedy
// slot iterations fused in-kernel. Similarity GEMV runs on the matrix pipe
// via V_WMMA_F32_16X16X4_F32 (exact fp32), selected-vector chunks fed from LDS.

typedef __attribute__((ext_vector_type(2))) float v2f;
typedef __attribute__((ext_vector_type(8))) float v8f;

#define N_TOK 1024
#define DIM   768
#define NSLOT 16
#define TPB   256      // threads per block
#define NWAVE 8        // TPB / 32
#define KCH   (DIM / 4)      // 192 WMMA K-chunks per 16-row group
#define NGRP  (N_TOK / 16)   // 64 row groups

__global__ __launch_bounds__(TPB)
void greedy_feature_init_kernel(const float* __restrict__ F,  // [B, N_TOK, DIM]
                                float* __restrict__ out) {    // [B, NSLOT, DIM]
    __shared__ float s_sal[N_TOK];     // saliency = ||f_n||
    __shared__ float s_inv[N_TOK];     // 1 / max(||f_n||, 1e-12)
    __shared__ float s_mask[N_TOK];    // suppression mask
    __shared__ float s_w[N_TOK];       // aggregation weights (unnormalized)
    __shared__ float s_sel[DIM];       // normalized selected vector
    __shared__ float s_red[TPB];       // reduction scratch (values)
    __shared__ int   s_redi[TPB];      // reduction scratch (indices)

    const int b    = blockIdx.x;
    const int tid  = threadIdx.x;
    const int lane = tid & 31;
    const int wave = tid >> 5;
    const int half = lane >> 4;     // 0: K0/K1 half, 1: K2/K3 half
    const int mrow = lane & 15;     // A-matrix row within the 16-row group

    const float* Fb = F + (size_t)b * N_TOK * DIM;

    // ---------------- setup: per-token norms, mask init --------------------
    for (int n = tid; n < N_TOK; n += TPB) {
        const float* row = Fb + (size_t)n * DIM;
        float s = 0.0f;
        #pragma unroll 4
        for (int k = 0; k < DIM; k += 4) {
            float4 v = *(const float4*)(row + k);
            s += v.x * v.x + v.y * v.y + v.z * v.z + v.w * v.w;
        }
        float nrm = sqrtf(s);
        s_sal[n]  = nrm;
        s_inv[n]  = 1.0f / fmaxf(nrm, 1e-12f);
        s_mask[n] = 1.0f;
    }
    __syncthreads();

    // ---------------- 16 sequential greedy slot iterations -----------------
    for (int slot = 0; slot < NSLOT; ++slot) {
        // --- argmax of saliency*mask (first-index wins on ties) ---
        float bv = -1.0f;
        int   bi = 0;
        #pragma unroll
        for (int j = 0; j < N_TOK / TPB; ++j) {
            int n = tid + j * TPB;                 // per-thread n is increasing
            float v = s_sal[n] * s_mask[n];
            if (v > bv) { bv = v; bi = n; }
        }
        s_red[tid] = bv; s_redi[tid] = bi;
        __syncthreads();
        for (int s = TPB >> 1; s > 0; s >>= 1) {
            if (tid < s) {
                float v2 = s_red[tid + s]; int i2 = s_redi[tid + s];
                float v1 = s_red[tid];     int i1 = s_redi[tid];
                if (v2 > v1 || (v2 == v1 && i2 < i1)) { s_red[tid] = v2; s_redi[tid] = i2; }
            }
            __syncthreads();
        }
        const int idx = s_redi[0];
        __syncthreads();   // everyone has idx; s_red reusable

        // --- build normalized selected vector in LDS ---
        const float* selrow = Fb + (size_t)idx * DIM;
        float vals[DIM / TPB];
        float part = 0.0f;
        #pragma unroll
        for (int j = 0; j < DIM / TPB; ++j) {
            float v = selrow[tid + j * TPB];
            vals[j] = v;
            part += v * v;
        }
        s_red[tid] = part;
        __syncthreads();
        for (int s = TPB >> 1; s > 0; s >>= 1) {
            if (tid < s) s_red[tid] += s_red[tid + s];
            __syncthreads();
        }
        const float selinv = 1.0f / fmaxf(sqrtf(s_red[0]), 1e-12f);
        #pragma unroll
        for (int j = 0; j < DIM / TPB; ++j)
            s_sel[tid + j * TPB] = vals[j] * selinv;
        __syncthreads();

        // --- phase 1: similarity GEMV on the matrix pipe ---
        // D(16x16) += A(16x4 fp32 rows of F) * B(4x16 = sel chunk in every col)
        for (int g = wave; g < NGRP; g += NWAVE) {
            const int rowbase = g << 4;
            // A layout (32-bit 16x4): lanes 0-15 -> K0,K1; lanes 16-31 -> K2,K3
            const float* arow = Fb + (size_t)(rowbase + mrow) * DIM + (half ? 2 : 0);
            const float* bsel = s_sel + (half ? 2 : 0);
            v8f acc = {};
            #pragma unroll 4
            for (int k0 = 0; k0 < DIM; k0 += 4) {
                v2f a = *(const v2f*)(arow + k0);   // global_load_b64
                v2f bb = *(const v2f*)(bsel + k0);  // ds_load_b64
                acc = __builtin_amdgcn_wmma_f32_16x16x4_f32(
                    false, a, false, bb, (short)0, acc, false, false);
            }
            // every column identical: lane with mrow==0 in each half owns 8 rows
            if (mrow == 0) {
                #pragma unroll
                for (int r = 0; r < 8; ++r) {
                    int n = rowbase + (half ? 8 : 0) + r;
                    float sim = acc[r] * s_inv[n];
                    float m = s_mask[n];
                    float w = (sim > 0.5f) ? fmaxf(sim * m, 0.0f) : 0.0f;
                    s_w[n] = w;
                    s_mask[n] = m * (1.0f - fminf(fmaxf(sim, 0.0f), 1.0f));
                }
            }
        }
        __syncthreads();

        // --- weight-sum reduction ---
        float ws = 0.0f;
        #pragma unroll
        for (int j = 0; j < N_TOK / TPB; ++j) ws += s_w[tid + j * TPB];
        s_red[tid] = ws;
        __syncthreads();
        for (int s = TPB >> 1; s > 0; s >>= 1) {
            if (tid < s) s_red[tid] += s_red[tid + s];
            __syncthreads();
        }
        const float rw = 1.0f / (s_red[0] + 1e-8f);

        // --- phase 2: slot = (sum_n w_n * f_n) * rw, coalesced over columns ---
        float acc0 = 0.0f, acc1 = 0.0f, acc2 = 0.0f;
        for (int n = 0; n < N_TOK; ++n) {
            float w = s_w[n];                       // wave-uniform broadcast
            if (w != 0.0f) {                        // uniform skip of zero rows
                const float* row = Fb + (size_t)n * DIM;
                acc0 += w * row[tid];
                acc1 += w * row[tid + TPB];
                acc2 += w * row[tid + 2 * TPB];
            }
        }
        float* op = out + ((size_t)b * NSLOT + slot) * DIM;
        op[tid]           = acc0 * rw;
        op[tid + TPB]     = acc1 * rw;
        op[tid + 2 * TPB] = acc2 * rw;
        __syncthreads();   // protect s_w / s_red / s_mask for next iteration
    }
}

extern "C" void kernel_launch(void* const* d_in, const int* in_sizes, int n_in,
                              void* d_out, int out_size, void* d_ws, size_t ws_size,
                              hipStream_t stream) {
    (void)n_in; (void)out_size; (void)d_ws; (void)ws_size;
    const float* features = (const float*)d_in[0];
    float* out = (float*)d_out;
    const int B = in_sizes[0] / (N_TOK * DIM);   // 64 for the reference shapes
    greedy_feature_init_kernel<<<B, TPB, 0, stream>>>(features, out);
}